// AttentionPooling_21466246545469
// MI455X (gfx1250) — compile-verified
//
#include <hip/hip_runtime.h>

// Attention pooling (segment softmax + weighted scatter-add), MI455X / gfx1250.
//
// Roofline: x = 1M x 128 fp32 = 512 MB dominates; HBM floor at 23.3 TB/s ~= 22 us.
// Strategy: one block per graph segment (batch is sorted), streaming online-softmax
// over 512-node chunks so all intermediates (scores, exp-weights, running max/sum,
// partial pooled vectors) live in LDS/registers. x is read twice per chunk but the
// second read hits L2 (chunk = 256 KB << 192 MB L2), so HBM traffic ~= one pass.
// Scores (GEMV X.q) are computed with V_WMMA_F32_16X16X4_F32 (16 nodes/wave, K
// chained over 128); pooling uses coalesced float4 loads + register accumulators.

typedef float v2f __attribute__((ext_vector_type(2)));
typedef float v8f __attribute__((ext_vector_type(8)));

#define DIM 128
#define NUM_GRAPHS 4096
#define CHUNK 512
#define BLOCK 256
#define NWAVES 8  // BLOCK / 32 (wave32)

__launch_bounds__(BLOCK)
__global__ void attn_pool_kernel(const float* __restrict__ x,
                                 const float* __restrict__ query,
                                 const int* __restrict__ batch,
                                 float* __restrict__ out,
                                 int n_nodes) {
  __shared__ float lds_q[DIM];                       // query vector
  __shared__ __align__(16) float lds_sc[CHUNK + 16]; // chunk scores / exp-weights (+tail pad)
  __shared__ float lds_red[BLOCK];                   // block reduction scratch
  __shared__ float lds_part[NWAVES][DIM];            // per-wave pooled partials
  __shared__ float lds_m, lds_scale, lds_s;

  const int tid  = threadIdx.x;
  const int lane = tid & 31;
  const int wave = tid >> 5;
  const int g    = blockIdx.x;

  if (tid < DIM) lds_q[tid] = query[tid];
  if (tid == 0) { lds_m = -3.402823466e38f; lds_s = 0.0f; lds_scale = 1.0f; }

  // Segment bounds via binary search in sorted batch (uniform -> scalarized loads).
  int lo = 0, hi = n_nodes;
  while (lo < hi) { int mid = (lo + hi) >> 1; if (batch[mid] < g) lo = mid + 1; else hi = mid; }
  const int seg_start = lo;
  hi = n_nodes;
  while (lo < hi) { int mid = (lo + hi) >> 1; if (batch[mid] < g + 1) lo = mid + 1; else hi = mid; }
  const int seg_end = lo;
  const int seg_len = seg_end - seg_start;

  // Per-wave pooled accumulator: lane l owns columns [4l, 4l+4).
  float acc0 = 0.f, acc1 = 0.f, acc2 = 0.f, acc3 = 0.f;

  __syncthreads();

  if (seg_len > 0) {
    for (int cbase = 0; cbase < seg_len; cbase += CHUNK) {
      const int clen = min(CHUNK, seg_len - cbase);

      // ---- phase 1: scores for this chunk via V_WMMA_F32_16X16X4_F32 ----
      // Wave computes 16 node-scores per group. A layout (16x4 f32): lanes 0-15
      // hold row M=lane with K={0,1} in VGPR0/1; lanes 16-31 hold K={2,3}.
      // B = query chunk broadcast into all 16 columns -> every D column equals
      // the score vector; chain C over k0 = 0..124 step 4.
      for (int i = wave * 16; i < clen; i += NWAVES * 16) {   // wave-uniform bound
        const int row  = lane & 15;
        const int koff = (lane >> 4) << 1;                    // 0 or 2
        int n = seg_start + cbase + i + row;
        n = min(n, seg_end - 1);                              // clamp tail rows
        const float* xp = x + (size_t)n * DIM + koff;
        v8f c = {};
        #pragma unroll 4
        for (int k0 = 0; k0 < DIM; k0 += 4) {
          v2f a = *(const v2f*)(xp + k0);
          v2f b = *(const v2f*)(&lds_q[k0 + koff]);
          c = __builtin_amdgcn_wmma_f32_16x16x4_f32(false, a, false, b,
                                                    (short)0, c, false, false);
        }
        // D layout: VGPR r, lanes 0-15 = D[r, lane]; lanes 16-31 = D[r+8, lane-16].
        // Column 0 suffices: lane 0 holds scores i..i+7 in c[0..7], lane 16 holds
        // scores i+8..i+15. Unguarded tail writes land past clen, below CHUNK (never read).
        if ((lane & 15) == 0) {
          const int base = i + ((lane >> 4) << 3);            // i or i+8 (16B aligned)
          float4* p = (float4*)&lds_sc[base];
          p[0] = make_float4(c[0], c[1], c[2], c[3]);         // ds_store_b128
          p[1] = make_float4(c[4], c[5], c[6], c[7]);         // ds_store_b128
        }
      }
      __syncthreads();

      // ---- phase 2: chunk max, fold into running max (online softmax) ----
      float lmax = -3.402823466e38f;
      for (int j = tid; j < clen; j += BLOCK) lmax = fmaxf(lmax, lds_sc[j]);
      lds_red[tid] = lmax;
      __syncthreads();
      for (int off = BLOCK / 2; off > 0; off >>= 1) {
        if (tid < off) lds_red[tid] = fmaxf(lds_red[tid], lds_red[tid + off]);
        __syncthreads();
      }
      if (tid == 0) {
        const float old_m = lds_m;
        const float new_m = fmaxf(old_m, lds_red[0]);
        const float scale = __expf(old_m - new_m);
        lds_s *= scale;
        lds_m = new_m;
        lds_scale = scale;
      }
      __syncthreads();
      const float new_m = lds_m;
      const float scale = lds_scale;
      acc0 *= scale; acc1 *= scale; acc2 *= scale; acc3 *= scale;

      // ---- phase 3: e = exp(score - m) in LDS, running sum ----
      float lsum = 0.f;
      for (int j = tid; j < clen; j += BLOCK) {
        const float e = __expf(lds_sc[j] - new_m);
        lds_sc[j] = e;
        lsum += e;
      }
      lds_red[tid] = lsum;
      __syncthreads();
      for (int off = BLOCK / 2; off > 0; off >>= 1) {
        if (tid < off) lds_red[tid] += lds_red[tid + off];
        __syncthreads();
      }
      if (tid == 0) lds_s += lds_red[0];
      __syncthreads();

      // ---- phase 4: weighted accumulate, one node per wave per step ----
      // Lane l loads float4 of cols [4l,4l+4): 512 B fully-coalesced per wave,
      // re-hitting the chunk just brought into L2 by phase 1.
      for (int j = wave; j < clen; j += NWAVES) {
        const float e = lds_sc[j];                  // LDS broadcast
        const float* xp = x + (size_t)(seg_start + cbase + j) * DIM + lane * 4;
        const float4 v = *(const float4*)xp;
        acc0 += e * v.x; acc1 += e * v.y; acc2 += e * v.z; acc3 += e * v.w;
      }
      __syncthreads();  // lds_sc reused next chunk
    }
  }

  // ---- final: merge 8 wave partials, normalize by running sum, store ----
  lds_part[wave][lane * 4 + 0] = acc0;
  lds_part[wave][lane * 4 + 1] = acc1;
  lds_part[wave][lane * 4 + 2] = acc2;
  lds_part[wave][lane * 4 + 3] = acc3;
  __syncthreads();
  if (tid < DIM) {
    float sum = 0.f;
    #pragma unroll
    for (int w = 0; w < NWAVES; ++w) sum += lds_part[w][tid];
    out[(size_t)g * DIM + tid] = (seg_len > 0) ? sum / lds_s : 0.0f;
  }
}

extern "C" void kernel_launch(void* const* d_in, const int* in_sizes, int n_in,
                              void* d_out, int out_size, void* d_ws, size_t ws_size,
                              hipStream_t stream) {
  const float* x     = (const float*)d_in[0];
  const float* query = (const float*)d_in[1];
  const int*   batch = (const int*)d_in[2];
  float* out = (float*)d_out;
  const int n_nodes = in_sizes[0] / DIM;  // 1,000,000
  (void)n_in; (void)out_size; (void)d_ws; (void)ws_size;
  attn_pool_kernel<<<NUM_GRAPHS, BLOCK, 0, stream>>>(x, query, batch, out, n_nodes);
}